// QuantumLayer_81028853007042
// MI455X (gfx1250) — compile-verified
//
#include <hip/hip_runtime.h>
#include <math.h>

// 4-qubit variational circuit, B = 1M samples.
// state-after-data-RYs is a rank-1 product state built in registers;
// the 3 weight layers + ring CNOTs collapse into one fixed 16x16 matrix M
// (built once per call in d_ws); the batched [B,16]x[16,16] matvec runs on
// v_wmma_f32_16x16x4_f32 (one 16-sample tile per wave, K chained 4x).

#define DIM 16

typedef float v2f __attribute__((ext_vector_type(2)));
typedef float v8f __attribute__((ext_vector_type(8)));

__host__ __device__ constexpr int cnot_perm_idx(int c, int t, int idx) {
    return idx ^ ((((idx >> (3 - c)) & 1)) << (3 - t));
}
__host__ __device__ constexpr int ring_perm_idx(int j) {
    // RING_PERM[j] = p0(p1(p2(p3(j)))), p_i = cnot(i, (i+1)%4)
    int v = cnot_perm_idx(3, 0, j);
    v = cnot_perm_idx(2, 3, v);
    v = cnot_perm_idx(1, 2, v);
    v = cnot_perm_idx(0, 1, v);
    return v;
}

// Build the fixed 16x16 operator M (final = M @ s) from the 12 weights.
// One thread per column; RY mixes rows (column-local), perm permutes rows.
__global__ void build_fixed_matrix_kernel(const float* __restrict__ w,
                                          float* __restrict__ M) {
    const int t = threadIdx.x;
    if (t >= DIM) return;
    float v[DIM];
#pragma unroll
    for (int i = 0; i < DIM; ++i) v[i] = (i == t) ? 1.0f : 0.0f;
#pragma unroll
    for (int layer = 0; layer < 3; ++layer) {
#pragma unroll
        for (int q = 0; q < 4; ++q) {
            const float th = 0.5f * w[layer * 4 + q];
            const float c = cosf(th);
            const float s = sinf(th);
            const int mask = 1 << (3 - q);
#pragma unroll
            for (int i = 0; i < DIM; ++i) {
                if ((i & mask) == 0) {
                    const int j = i | mask;
                    const float a0 = v[i], a1 = v[j];
                    v[i] = c * a0 - s * a1;   // bit=0 row
                    v[j] = s * a0 + c * a1;   // bit=1 row
                }
            }
        }
        float nv[DIM];
#pragma unroll
        for (int i = 0; i < DIM; ++i) nv[i] = v[ring_perm_idx(i)];
#pragma unroll
        for (int i = 0; i < DIM; ++i) v[i] = nv[i];
    }
#pragma unroll
    for (int i = 0; i < DIM; ++i) M[i * DIM + t] = v[i];   // row-major M
}

// Main kernel: 8 waves/block, one 16-sample tile per wave.
__global__ void __launch_bounds__(256)
quantum_layer_kernel(const float* __restrict__ x, const float* __restrict__ M,
                     float* __restrict__ out, int ntiles, long Btot) {
    __shared__ float probs[8][DIM * 17];          // padded 16x16 tile per wave
    const int lane = threadIdx.x & 31;
    const int wave = threadIdx.x >> 5;
    const int m    = lane & 15;                   // A: sample row / B: matrix row
    const int h    = lane >> 4;                   // K-half selector per ISA layout
    const long tile = (long)blockIdx.x * 8 + wave;
    const bool active = tile < (long)ntiles;      // wave-uniform -> EXEC all 1s

    if (active) {
        long b = tile * 16 + m;
        const long bl = (b < Btot) ? b : (Btot - 1);
        const float4 xv = ((const float4*)x)[bl];

        float c0, s0, c1, s1, c2, s2, c3, s3;
        sincosf(0.5f * xv.x, &s0, &c0);
        sincosf(0.5f * xv.y, &s1, &c1);
        sincosf(0.5f * xv.z, &s2, &c2);
        sincosf(0.5f * xv.w, &s3, &c3);
        const float f2 = h ? s2 : c2;             // qubit-2 factor (bit1 of k == h)

        v8f acc = {0.f, 0.f, 0.f, 0.f, 0.f, 0.f, 0.f, 0.f};
#pragma unroll
        for (int g = 0; g < 4; ++g) {
            // A operand: S[m, k] with k = 4g + 2h + e  (product-state amplitude)
            // k bits: b3=(g>>1), b2=(g&1), b1=h, b0=e ; factor f_q selected by b_{3-q}
            const float base = ((g & 2) ? s0 : c0) * ((g & 1) ? s1 : c1) * f2;
            v2f a;
            a.x = base * c3;                      // e = 0
            a.y = base * s3;                      // e = 1
            // B operand: B[k][n] = M[n][k], n = m (lane&15), same k pattern
            const float* mrow = M + m * DIM + g * 4 + 2 * h;
            v2f bm;
            bm.x = mrow[0];
            bm.y = mrow[1];
            acc = __builtin_amdgcn_wmma_f32_16x16x4_f32(
                false, a, false, bm, (short)0, acc, false, false);
        }
        // D layout: VGPR r holds Y[row = r + 8h, col = m]; store probabilities
#pragma unroll
        for (int r = 0; r < 8; ++r) {
            const int row = r + 8 * h;
            const float yv = acc[r];
            probs[wave][row * 17 + m] = yv * yv;
        }
    }
    __syncthreads();
    if (active) {
        const long b = tile * 16 + m;             // this lane's sample row
        if (b < Btot) {
            const int bp0 = 3 - (2 * h);          // q = 2h   -> bit (3-q)
            const int bp1 = 3 - (2 * h + 1);      // q = 2h+1
            float a0 = 0.f, a1 = 0.f;
#pragma unroll
            for (int n = 0; n < DIM; ++n) {
                const float p = probs[wave][m * 17 + n];
                a0 += ((n >> bp0) & 1) ? -p : p;
                a1 += ((n >> bp1) & 1) ? -p : p;
            }
            float2 r2;
            r2.x = a0;
            r2.y = a1;
            ((float2*)out)[b * 2 + h] = r2;       // out[b][2h], out[b][2h+1]
        }
    }
}

extern "C" void kernel_launch(void* const* d_in, const int* in_sizes, int n_in,
                              void* d_out, int out_size, void* d_ws, size_t ws_size,
                              hipStream_t stream) {
    const float* x = (const float*)d_in[0];       // [B,4] float32
    const float* w = (const float*)d_in[1];       // [12]  float32
    float* out = (float*)d_out;                   // [B,4] float32
    float* M = (float*)d_ws;                      // 16x16 scratch

    const long B = (long)in_sizes[0] / 4;
    const int ntiles = (int)((B + 15) / 16);
    const int blocks = (ntiles + 7) / 8;

    build_fixed_matrix_kernel<<<1, 16, 0, stream>>>(w, M);
    quantum_layer_kernel<<<blocks, 256, 0, stream>>>(x, M, out, ntiles, B);
}